// Embedding_17815524344037
// MI455X (gfx1250) — compile-verified
//
#include <hip/hip_runtime.h>
#include <hip/hip_bf16.h>

typedef __attribute__((ext_vector_type(2))) float v2f;
typedef __attribute__((ext_vector_type(8))) float v8f;

#define WMMA4(acc, a, b) \
  acc = __builtin_amdgcn_wmma_f32_16x16x4_f32(false, (a), false, (b), (short)0, (acc), false, false)

// ---- model constants ----
#define FDIM 128
#define RDIM 32
#define MU0   0.0067379470f          // exp(-5)
#define DMU   0.0320407110f          // (1-exp(-5))/31
#define BETA  259.48498f             // (2/32*(1-exp(-5)))^-2
#define PI_OVER_C 0.62831853071f     // pi/5

// ---- workspace float offsets ----
#define WS_FUSED_RAW   0           // 144*128
#define WS_PACK_FUSED  18432       // 8nt*36c*32lane*2
#define WS_PACK_F0     36864       // 3w*8nt*8c*32lane*2
#define WS_PACK_AFTER  49152       // 3w*8nt*32c*32lane*2
#define WS_XDEC        98304       // N*9*128

__device__ __forceinline__ v8f bcast8(float b) {
  v8f r = {b, b, b, b, b, b, b, b};
  return r;
}
__device__ __forceinline__ float silu_f(float x) {
  return x * (1.0f / (1.0f + __expf(-x)));
}

// ============ Kernel 0: fuse W_node/W_edge into W_cat (144x128 fused K) ============
__global__ void k_prep_fused(const float* __restrict__ Wn, const float* __restrict__ We,
                             const float* __restrict__ Wc, float* __restrict__ WfRaw) {
  int idx = blockIdx.x * blockDim.x + threadIdx.x;
  if (idx >= 144 * 128) return;
  int k = idx >> 7, g = idx & 127;
  float acc = 0.0f;
  if (k < 64) {
    for (int j = 0; j < 128; ++j) acc += Wn[k * 128 + j] * Wc[j * 128 + g];
  } else if (k < 128) {
    int kk = k - 64;
    for (int j = 0; j < 128; ++j) acc += Wn[kk * 128 + j] * Wc[(128 + j) * 128 + g];
  } else {
    int kk = k - 128;
    for (int j = 0; j < 128; ++j) acc += We[kk * 128 + j] * Wc[(256 + j) * 128 + g];
  }
  WfRaw[idx] = acc;
}

// ============ Kernel 1: swizzle all B matrices into WMMA fragment layout ============
// B fragment for a 4-K chunk: lane (h=lane>>4, n=lane&15) holds {B[4c+2h][col], B[4c+2h+1][col]}
__global__ void k_pack(const float* __restrict__ WfRaw,
                       const float* __restrict__ WI, const float* __restrict__ WA,
                       const float* __restrict__ WS_,
                       const float* __restrict__ WIa, const float* __restrict__ WAa,
                       const float* __restrict__ WSa, float* __restrict__ ws) {
  int idx = blockIdx.x * blockDim.x + threadIdx.x;
  if (idx < 9216) {  // fused edge weights: (nt*36+c)*32+lane
    int lane = idx & 31, rest = idx >> 5;
    int c = rest % 36, nt = rest / 36;
    int h = lane >> 4, n = lane & 15;
    int k = 4 * c + 2 * h, col = nt * 16 + n;
    ws[WS_PACK_FUSED + idx * 2 + 0] = WfRaw[k * 128 + col];
    ws[WS_PACK_FUSED + idx * 2 + 1] = WfRaw[(k + 1) * 128 + col];
  } else if (idx < 9216 + 6144) {  // radial weights: ((w*8+nt)*8+c)*32+lane
    int i = idx - 9216;
    int lane = i & 31, rest = i >> 5;
    int c = rest & 7, rest2 = rest >> 3;
    int nt = rest2 & 7, w = rest2 >> 3;
    const float* W = (w == 0) ? WI : ((w == 1) ? WA : WS_);
    int h = lane >> 4, n = lane & 15;
    int k = 4 * c + 2 * h, col = nt * 16 + n;
    ws[WS_PACK_F0 + i * 2 + 0] = W[k * 128 + col];
    ws[WS_PACK_F0 + i * 2 + 1] = W[(k + 1) * 128 + col];
  } else if (idx < 9216 + 6144 + 24576) {  // after weights: w*8192+(nt*32+c)*32+lane
    int i = idx - 15360;
    int lane = i & 31, rest = i >> 5;
    int c = rest & 31, rest2 = rest >> 5;
    int nt = rest2 & 7, w = rest2 >> 3;
    const float* W = (w == 0) ? WIa : ((w == 1) ? WAa : WSa);
    int h = lane >> 4, n = lane & 15;
    int k = 4 * c + 2 * h, col = nt * 16 + n;
    ws[WS_PACK_AFTER + i * 2 + 0] = W[k * 128 + col];
    ws[WS_PACK_AFTER + i * 2 + 1] = W[(k + 1) * 128 + col];
  }
}

// ============ Kernel 2: zero decomposed accumulator ============
__global__ void k_zero(float* __restrict__ p, size_t n) {
  size_t i = (size_t)blockIdx.x * blockDim.x + threadIdx.x;
  size_t stride = (size_t)gridDim.x * blockDim.x;
  for (; i < n; i += stride) p[i] = 0.0f;
}

// ============ Kernel 3: per-edge WMMA + decomposed atomic scatter ============
// One wave32 per 16-edge tile. D layout: VGPR r -> row m=r+8h, col n=lane&15.
__global__ __launch_bounds__(256) void k_edges(
    const float* __restrict__ coords, const float* __restrict__ nodeA,
    const float* __restrict__ edgeA, const float* __restrict__ bcat,
    const float* __restrict__ bI, const float* __restrict__ bA,
    const float* __restrict__ bS, const int* __restrict__ ei,
    const float* __restrict__ packFused, const float* __restrict__ packF0,
    float* __restrict__ Xdec, int E, int nTiles) {
  int lane = threadIdx.x & 31;
  int tile = blockIdx.x * 8 + (threadIdx.x >> 5);
  if (tile >= nTiles) return;  // wave-uniform: EXEC stays all-ones for WMMA
  int m = lane & 15, h = lane >> 4;
  int e = tile * 16 + m;
  int snd = ei[e], rcv = ei[E + e];

  // geometry (duplicated across the two lane halves)
  float dx = coords[rcv * 3 + 0] - coords[snd * 3 + 0];
  float dy = coords[rcv * 3 + 1] - coords[snd * 3 + 1];
  float dz = coords[rcv * 3 + 2] - coords[snd * 3 + 2];
  float len = sqrtf(dx * dx + dy * dy + dz * dz);
  float inv = 1.0f / len;
  float ux = dx * inv, uy = dy * inv, uz = dz * inv;
  float phi = (len < 5.0f) ? 0.5f * (__cosf(len * PI_OVER_C) + 1.0f) : 0.0f;
  float elen = __expf(-len);

  // rp = rbf*phi A-fragments (M=16 edges, K=32): lane holds K=4c+2h, 4c+2h+1
  v2f rpf[8];
#pragma unroll
  for (int c = 0; c < 8; ++c) {
    int k0 = 4 * c + 2 * h;
    float d0 = elen - (MU0 + (float)k0 * DMU);
    float d1 = elen - (MU0 + (float)(k0 + 1) * DMU);
    rpf[c].x = __expf(-BETA * d0 * d0) * phi;
    rpf[c].y = __expf(-BETA * d1 * d1) * phi;
  }

  // per-output-row geometry via wave32 shuffles (row m = r + 8h)
  float ux_r[8], uy_r[8], uz_r[8], phi_r[8];
  int rcv_r[8];
#pragma unroll
  for (int r = 0; r < 8; ++r) {
    int src = r + 8 * h;
    ux_r[r] = __shfl(ux, src);
    uy_r[r] = __shfl(uy, src);
    uz_r[r] = __shfl(uz, src);
    phi_r[r] = __shfl(phi, src);
    rcv_r[r] = __shfl(rcv, src);
  }

  // fused feature GEMM: cat @ W_cat folded to K=144 over [attrs[rcv]|attrs[snd]|edge_attrs]
  const float* rowR = nodeA + (size_t)rcv * 64;
  const float* rowS = nodeA + (size_t)snd * 64;
  const float* rowE = edgeA + (size_t)e * 16;
  v8f feat[8];
#pragma unroll
  for (int nt = 0; nt < 8; ++nt) feat[nt] = bcast8(bcat[nt * 16 + m]);
#pragma unroll
  for (int c = 0; c < 36; ++c) {
    int k = 4 * c + 2 * h;
    v2f a;
    if (4 * c < 64)       a = *(const v2f*)(rowR + k);
    else if (4 * c < 128) a = *(const v2f*)(rowS + (k - 64));
    else                  a = *(const v2f*)(rowE + (k - 128));
#pragma unroll
    for (int nt = 0; nt < 8; ++nt) {
      v2f b = *(const v2f*)(packFused + ((size_t)((nt * 36 + c) * 32 + lane)) * 2);
      WMMA4(feat[nt], a, b);
    }
  }

  // three radial GEMMs (share rp A-frags) + decomposed scatter
#pragma unroll
  for (int w = 0; w < 3; ++w) {
    const float* bias = (w == 0) ? bI : ((w == 1) ? bA : bS);
    v8f facc[8];
#pragma unroll
    for (int nt = 0; nt < 8; ++nt) facc[nt] = bcast8(bias[nt * 16 + m]);
#pragma unroll
    for (int c = 0; c < 8; ++c) {
#pragma unroll
      for (int nt = 0; nt < 8; ++nt) {
        v2f b = *(const v2f*)(packF0 + ((size_t)(((w * 8 + nt) * 8 + c) * 32 + lane)) * 2);
        WMMA4(facc[nt], rpf[c], b);
      }
    }
#pragma unroll
    for (int nt = 0; nt < 8; ++nt) {
#pragma unroll
      for (int r = 0; r < 8; ++r) {
        float f0 = facc[nt][r] * phi_r[r];         // (rp@W + b) * phi
        float g = feat[nt][r] * f0;                // g_{I,A,S}
        float* base = Xdec + (size_t)rcv_r[r] * 1152 + (nt * 16 + m);
        if (w == 0) {
          unsafeAtomicAdd(base + 0 * 128, g);      // trace part
        } else if (w == 1) {
          unsafeAtomicAdd(base + 1 * 128, g * ux_r[r]);
          unsafeAtomicAdd(base + 2 * 128, g * uy_r[r]);
          unsafeAtomicAdd(base + 3 * 128, g * uz_r[r]);
        } else {
          float x = ux_r[r], y = uy_r[r], z = uz_r[r];
          unsafeAtomicAdd(base + 4 * 128, g * (x * x - (1.0f / 3.0f)));
          unsafeAtomicAdd(base + 5 * 128, g * (y * y - (1.0f / 3.0f)));
          unsafeAtomicAdd(base + 6 * 128, g * (x * y));
          unsafeAtomicAdd(base + 7 * 128, g * (x * z));
          unsafeAtomicAdd(base + 8 * 128, g * (y * z));
        }
      }
    }
  }
}

// ============ Kernel 4: per-node norm -> LayerNorm -> MLP gates ============
__global__ __launch_bounds__(128) void k_node_mlp(
    const float* __restrict__ Xdec, const float* __restrict__ gamma,
    const float* __restrict__ beta, const float* __restrict__ W1,
    const float* __restrict__ b1, const float* __restrict__ W2,
    const float* __restrict__ b2, float* __restrict__ fias) {
  int n = blockIdx.x, c = threadIdx.x;
  const float* xb = Xdec + (size_t)n * 1152;
  float t = xb[c];
  float ax = xb[128 + c], ay = xb[256 + c], az = xb[384 + c];
  float s00 = xb[512 + c], s11 = xb[640 + c];
  float s01 = xb[768 + c], s02 = xb[896 + c], s12 = xb[1024 + c];
  float s22 = -(s00 + s11);
  // ||X_i||_F^2 from orthogonal decomposition
  float xn = 3.0f * t * t + 2.0f * (ax * ax + ay * ay + az * az) +
             s00 * s00 + s11 * s11 + s22 * s22 +
             2.0f * (s01 * s01 + s02 * s02 + s12 * s12);

  __shared__ float s1[128], s2[128], hb[128], h1[128];
  s1[c] = xn;
  s2[c] = xn * xn;
  __syncthreads();
  for (int s = 64; s > 0; s >>= 1) {
    if (c < s) { s1[c] += s1[c + s]; s2[c] += s2[c + s]; }
    __syncthreads();
  }
  float mean = s1[0] * (1.0f / 128.0f);
  float var = s2[0] * (1.0f / 128.0f) - mean * mean;
  float hv = (xn - mean) * rsqrtf(var + 1e-5f) * gamma[c] + beta[c];
  hb[c] = hv;
  __syncthreads();

  float acc = b1[c];
  for (int k = 0; k < 128; ++k) acc += hb[k] * W1[k * 128 + c];
  h1[c] = silu_f(acc);
  __syncthreads();

  float o0 = b2[3 * c], o1 = b2[3 * c + 1], o2 = b2[3 * c + 2];
  for (int k = 0; k < 128; ++k) {
    float hk = h1[k];
    const float* wr = W2 + (size_t)k * 384 + 3 * c;
    o0 += hk * wr[0];
    o1 += hk * wr[1];
    o2 += hk * wr[2];
  }
  fias[(size_t)n * 384 + c]       = silu_f(o0);  // f_I
  fias[(size_t)n * 384 + 128 + c] = silu_f(o1);  // f_A
  fias[(size_t)n * 384 + 256 + c] = silu_f(o2);  // f_S
}

// ============ Kernel 5: 9 channel-mixing WMMA GEMMs + output assembly ============
__global__ __launch_bounds__(256) void k_assemble(
    const float* __restrict__ Xdec, const float* __restrict__ packAfter,
    const float* __restrict__ fias, float* __restrict__ out, int nTiles) {
  int lane = threadIdx.x & 31;
  int tile = blockIdx.x * 8 + (threadIdx.x >> 5);
  if (tile >= nTiles) return;
  int n0 = tile * 16, m = lane & 15, h = lane >> 4;
  const float* arow = Xdec + (size_t)(n0 + m) * 1152;  // this lane's A-matrix row (node n0+m)

  for (int nt = 0; nt < 8; ++nt) {
    v8f d[9];
#pragma unroll
    for (int comp = 0; comp < 9; ++comp) {
      int w = (comp == 0) ? 0 : ((comp < 4) ? 1 : 2);  // W_I_after / W_A_after / W_S_after
      v8f acc = bcast8(0.0f);
      const float* bbase = packAfter + (size_t)w * 16384 + (size_t)nt * 2048 + lane * 2;
#pragma unroll
      for (int c = 0; c < 32; ++c) {
        v2f a = *(const v2f*)(arow + comp * 128 + 4 * c + 2 * h);
        v2f b = *(const v2f*)(bbase + c * 64);
        WMMA4(acc, a, b);
      }
      d[comp] = acc;
    }
#pragma unroll
    for (int r = 0; r < 8; ++r) {
      int node = n0 + r + 8 * h;
      int g = nt * 16 + m;
      const float* fb = fias + (size_t)node * 384;
      float fI = fb[g], fA = fb[128 + g], fS = fb[256 + g];
      float tv = d[0][r];
      float aX = d[1][r], aY = d[2][r], aZ = d[3][r];
      float s00 = d[4][r], s11 = d[5][r];
      float s01 = d[6][r], s02 = d[7][r], s12 = d[8][r];
      float s22 = -(s00 + s11);
      float* o = out + ((size_t)node * 128 + g) * 9;
      o[0] =  fI * tv + fS * s00;
      o[1] =  fA * aZ + fS * s01;
      o[2] = -fA * aY + fS * s02;
      o[3] = -fA * aZ + fS * s01;
      o[4] =  fI * tv + fS * s11;
      o[5] =  fA * aX + fS * s12;
      o[6] =  fA * aY + fS * s02;
      o[7] = -fA * aX + fS * s12;
      o[8] =  fI * tv + fS * s22;
    }
  }
}

extern "C" void kernel_launch(void* const* d_in, const int* in_sizes, int n_in,
                              void* d_out, int out_size, void* d_ws, size_t ws_size,
                              hipStream_t stream) {
  const float* coords = (const float*)d_in[0];
  const float* nodeA  = (const float*)d_in[1];
  const float* edgeA  = (const float*)d_in[2];
  const float* Wn     = (const float*)d_in[3];
  const float* We     = (const float*)d_in[4];
  const float* Wc     = (const float*)d_in[5];
  const float* bcat   = (const float*)d_in[6];
  const float* WI     = (const float*)d_in[7];
  const float* bI     = (const float*)d_in[8];
  const float* WA     = (const float*)d_in[9];
  const float* bA     = (const float*)d_in[10];
  const float* WS_    = (const float*)d_in[11];
  const float* bS     = (const float*)d_in[12];
  const float* gamma  = (const float*)d_in[13];
  const float* beta   = (const float*)d_in[14];
  const float* W1     = (const float*)d_in[15];
  const float* b1     = (const float*)d_in[16];
  const float* W2     = (const float*)d_in[17];
  const float* b2     = (const float*)d_in[18];
  const float* WIa    = (const float*)d_in[19];
  const float* WAa    = (const float*)d_in[20];
  const float* WSa    = (const float*)d_in[21];
  const int*   ei     = (const int*)d_in[22];

  int N = in_sizes[0] / 3;
  int E = in_sizes[22] / 2;

  float* ws = (float*)d_ws;
  float* WfRaw     = ws + WS_FUSED_RAW;
  float* packFused = ws + WS_PACK_FUSED;
  float* packF0    = ws + WS_PACK_F0;
  float* packAfter = ws + WS_PACK_AFTER;
  float* Xdec      = ws + WS_XDEC;
  float* fias      = Xdec + (size_t)N * 1152;
  float* outp      = (float*)d_out;

  k_prep_fused<<<(144 * 128 + 255) / 256, 256, 0, stream>>>(Wn, We, Wc, WfRaw);
  k_pack<<<(9216 + 6144 + 24576 + 255) / 256, 256, 0, stream>>>(
      WfRaw, WI, WA, WS_, WIa, WAa, WSa, ws);
  k_zero<<<2048, 256, 0, stream>>>(Xdec, (size_t)N * 1152);

  int eTiles = E / 16;
  k_edges<<<(eTiles + 7) / 8, 256, 0, stream>>>(
      coords, nodeA, edgeA, bcat, bI, bA, bS, ei, packFused, packF0, Xdec, E, eTiles);

  k_node_mlp<<<N, 128, 0, stream>>>(Xdec, gamma, beta, W1, b1, W2, b2, fias);

  int nTiles = N / 16;
  k_assemble<<<(nTiles + 7) / 8, 256, 0, stream>>>(Xdec, packAfter, fias, outp, nTiles);
}